// SoftPoolFeat_13812614824492
// MI455X (gfx1250) — compile-verified
//
#include <hip/hip_runtime.h>
#include <hip/hip_bf16.h>

typedef float v2f __attribute__((ext_vector_type(2)));
typedef float v8f __attribute__((ext_vector_type(8)));

#define BATCH 16
#define NPTS 8192
#define PKEEP 1024
#define REG 16
#define BN_EPS 1e-5f

// ---------------------------------------------------------------------------
// Kernel 1: fused point MLP 3->64->128->256 (+BN/ReLU) and scorer 256->16.
// One wave32 per 16-point tile. All GEMM layers run on v_wmma_f32_16x16x4_f32.
// ---------------------------------------------------------------------------
__global__ __launch_bounds__(64) void mlp_kernel(
    const float* __restrict__ x,
    const float* __restrict__ W1, const float* __restrict__ b1,
    const float* __restrict__ g1, const float* __restrict__ be1,
    const float* __restrict__ W2, const float* __restrict__ b2,
    const float* __restrict__ g2, const float* __restrict__ be2,
    const float* __restrict__ W3, const float* __restrict__ b3,
    const float* __restrict__ g3, const float* __restrict__ be3,
    const float* __restrict__ Ws, const float* __restrict__ bs,
    float* __restrict__ feat,   // [B,256,N]
    float* __restrict__ val)    // [B,16,N]
{
    const int lane = threadIdx.x & 31;
    const int wv   = threadIdx.x >> 5;           // wave in block (0..1)
    const int tile = blockIdx.x * 2 + wv;        // 16-point tile id
    const int tilesPerBatch = NPTS / 16;
    const int b  = tile / tilesPerBatch;
    const int n0 = (tile % tilesPerBatch) * 16;

    const int n  = lane & 15;   // column (point) within tile
    const int hi = lane >> 4;   // lane half

    // per-wave LDS staging tiles for C/D-layout -> B-layout redistribution
    __shared__ float h2tile[2][128][16];
    __shared__ float fttile[2][256][16];
    float (*h2t)[16] = h2tile[wv];
    float (*ftt)[16] = fttile[wv];

    const float invs = rsqrtf(1.0f + BN_EPS);

    // ---- Layer 1 (K=3): computed per-lane directly in WMMA B-operand layout.
    // Chunk c holds channels {4c+2*hi, 4c+2*hi+1} at column n.
    const float x0 = x[((long)b * 3 + 0) * NPTS + n0 + n];
    const float x1 = x[((long)b * 3 + 1) * NPTS + n0 + n];
    const float x2 = x[((long)b * 3 + 2) * NPTS + n0 + n];

    v2f h1b[16];
#pragma unroll
    for (int c = 0; c < 16; ++c) {
#pragma unroll
        for (int k = 0; k < 2; ++k) {
            const int ch = 4 * c + 2 * hi + k;
            float y = W1[ch * 3 + 0] * x0 + W1[ch * 3 + 1] * x1 + W1[ch * 3 + 2] * x2;
            y += b1[ch];
            y = y * (g1[ch] * invs) + be1[ch];
            y = fmaxf(y, 0.0f);
            if (k == 0) h1b[c].x = y; else h1b[c].y = y;
        }
    }

    // ---- Layer 2: h2[128x16] = relu(bn2(W2[128x64] * h1))
#pragma unroll 2
    for (int mt = 0; mt < 8; ++mt) {
        v8f acc = {};
        const int m = mt * 16 + (lane & 15);
#pragma unroll
        for (int c = 0; c < 16; ++c) {
            v2f a;
            a.x = W2[m * 64 + 4 * c + 2 * hi];
            a.y = W2[m * 64 + 4 * c + 2 * hi + 1];
            acc = __builtin_amdgcn_wmma_f32_16x16x4_f32(
                false, a, false, h1b[c], (short)0, acc, false, false);
        }
#pragma unroll
        for (int v = 0; v < 8; ++v) {
            const int ch = mt * 16 + v + 8 * hi;
            float y = acc[v] + b2[ch];
            y = y * (g2[ch] * invs) + be2[ch];
            h2t[ch][n] = fmaxf(y, 0.0f);
        }
    }

    // ---- Layer 3: feat[256x16] = bn3(W3[256x128] * h2)   (no relu)
#pragma unroll 1
    for (int mt = 0; mt < 16; ++mt) {
        v8f acc = {};
        const int m = mt * 16 + (lane & 15);
#pragma unroll
        for (int c = 0; c < 32; ++c) {
            v2f a, bf;
            a.x  = W3[m * 128 + 4 * c + 2 * hi];
            a.y  = W3[m * 128 + 4 * c + 2 * hi + 1];
            bf.x = h2t[4 * c + 2 * hi][n];
            bf.y = h2t[4 * c + 2 * hi + 1][n];
            acc = __builtin_amdgcn_wmma_f32_16x16x4_f32(
                false, a, false, bf, (short)0, acc, false, false);
        }
#pragma unroll
        for (int v = 0; v < 8; ++v) {
            const int ch = mt * 16 + v + 8 * hi;
            float y = acc[v] + b3[ch];
            y = y * (g3[ch] * invs) + be3[ch];
            ftt[ch][n] = y;
            feat[((long)b * 256 + ch) * NPTS + n0 + n] = y;
        }
    }

    // ---- Layer 4 (scorer): val[16x16] = Ws[16x256] * feat + bs
    {
        v8f acc = {};
        const int m = lane & 15;
#pragma unroll
        for (int c = 0; c < 64; ++c) {
            v2f a, bf;
            a.x  = Ws[m * 256 + 4 * c + 2 * hi];
            a.y  = Ws[m * 256 + 4 * c + 2 * hi + 1];
            bf.x = ftt[4 * c + 2 * hi][n];
            bf.y = ftt[4 * c + 2 * hi + 1][n];
            acc = __builtin_amdgcn_wmma_f32_16x16x4_f32(
                false, a, false, bf, (short)0, acc, false, false);
        }
#pragma unroll
        for (int v = 0; v < 8; ++v) {
            const int r = v + 8 * hi;
            val[((long)b * REG + r) * NPTS + n0 + n] = acc[v] + bs[r];
        }
    }
}

// ---------------------------------------------------------------------------
// Kernel 2: exact descending top-1024 per (b, region) via bitonic sort in LDS.
// ---------------------------------------------------------------------------
__global__ __launch_bounds__(256) void topk_kernel(
    const float* __restrict__ val, int* __restrict__ idxOut)
{
    __shared__ float sv[NPTS];
    __shared__ unsigned short si[NPTS];

    const float* v = val + (long)blockIdx.x * NPTS;
    for (int i = threadIdx.x; i < NPTS; i += 256) {
        sv[i] = v[i];
        si[i] = (unsigned short)i;
    }
    __syncthreads();

    for (int ksz = 2; ksz <= NPTS; ksz <<= 1) {
        for (int j = ksz >> 1; j > 0; j >>= 1) {
            for (int i = threadIdx.x; i < NPTS; i += 256) {
                const int l = i ^ j;
                if (l > i) {
                    const float a = sv[i], bb = sv[l];
                    const unsigned short ia = si[i], ib = si[l];
                    // "i ranks after l" in descending order (index tie-break)
                    const bool after = (a < bb) || (a == bb && ia > ib);
                    if (after == ((i & ksz) == 0)) {
                        sv[i] = bb; sv[l] = a;
                        si[i] = ib; si[l] = ia;
                    }
                }
            }
            __syncthreads();
        }
    }

    for (int i = threadIdx.x; i < PKEEP; i += 256)
        idxOut[(long)blockIdx.x * PKEEP + i] = (int)si[i];
}

// ---------------------------------------------------------------------------
// Kernel 3: gather sp_cube, cabin max-pool, sp_idx broadcast.
// One block per (b, region).
// ---------------------------------------------------------------------------
__global__ __launch_bounds__(256) void gather_kernel(
    const float* __restrict__ feat, const int* __restrict__ idx,
    float* __restrict__ out)
{
    const int br = blockIdx.x;
    const int b  = br >> 4;
    const int r  = br & 15;

    __shared__ int   sidx[PKEEP];
    __shared__ float gbuf[PKEEP];

    const int* ip = idx + (long)br * PKEEP;
    for (int i = threadIdx.x; i < PKEEP; i += 256) sidx[i] = ip[i];
    __syncthreads();

    float* sp_cube = out;                                          // [B,256,1,16384]
    float* cabins  = out + (size_t)BATCH * 256 * (REG * PKEEP);    // [B,256,16,8]
    float* sp_idx  = cabins + (size_t)BATCH * 256 * REG * 8;       // [B,19,1,16384]

    // sp_idx: indices as float, broadcast over 19 channels
    for (int c = 0; c < REG + 3; ++c)
        for (int i = threadIdx.x; i < PKEEP; i += 256)
            sp_idx[((long)b * (REG + 3) + c) * (REG * PKEEP) + r * PKEEP + i] =
                (float)sidx[i];

    // per channel: gather + cabin max (8 bins of 128 points)
    for (int c = 0; c < 256; ++c) {
        const float* f = feat + ((long)b * 256 + c) * NPTS;
        for (int i = threadIdx.x; i < PKEEP; i += 256) {
            const float v = f[sidx[i]];
            gbuf[i] = v;
            sp_cube[((long)b * 256 + c) * (REG * PKEEP) + r * PKEEP + i] = v;
        }
        __syncthreads();
        if (threadIdx.x < 8) {
            float m = gbuf[threadIdx.x * 128];
            for (int t = 1; t < 128; ++t)
                m = fmaxf(m, gbuf[threadIdx.x * 128 + t]);
            cabins[(((long)b * 256 + c) * REG + r) * 8 + threadIdx.x] = m;
        }
        __syncthreads();
    }
}

// ---------------------------------------------------------------------------
extern "C" void kernel_launch(void* const* d_in, const int* in_sizes, int n_in,
                              void* d_out, int out_size, void* d_ws, size_t ws_size,
                              hipStream_t stream) {
    const float* x   = (const float*)d_in[0];
    const float* W1  = (const float*)d_in[1];
    const float* b1  = (const float*)d_in[2];
    const float* g1  = (const float*)d_in[3];
    const float* be1 = (const float*)d_in[4];
    const float* W2  = (const float*)d_in[5];
    const float* b2  = (const float*)d_in[6];
    const float* g2  = (const float*)d_in[7];
    const float* be2 = (const float*)d_in[8];
    const float* W3  = (const float*)d_in[9];
    const float* b3  = (const float*)d_in[10];
    const float* g3  = (const float*)d_in[11];
    const float* be3 = (const float*)d_in[12];
    const float* Ws  = (const float*)d_in[13];
    const float* bs  = (const float*)d_in[14];
    float* out = (float*)d_out;

    // workspace layout
    float* feat = (float*)d_ws;                                   // B*256*N floats
    float* val  = feat + (size_t)BATCH * 256 * NPTS;              // B*16*N floats
    int*   idx  = (int*)(val + (size_t)BATCH * REG * NPTS);       // B*16*P ints

    const int tiles = BATCH * (NPTS / 16);   // 8192 wave-tiles, 2 per block
    mlp_kernel<<<tiles / 2, 64, 0, stream>>>(
        x, W1, b1, g1, be1, W2, b2, g2, be2, W3, b3, g3, be3, Ws, bs, feat, val);

    topk_kernel<<<BATCH * REG, 256, 0, stream>>>(val, idx);

    gather_kernel<<<BATCH * REG, 256, 0, stream>>>(feat, idx, out);
}